// SoftPhongNormalShader_36197984371117
// MI455X (gfx1250) — compile-verified
//
#include <hip/hip_runtime.h>

namespace {
constexpr float kSigmaInv  = 1.0f / 1e-4f;           // 1/SIGMA
constexpr float kGammaInv  = 1.0f / 1e-4f;           // 1/GAMMA
constexpr float kEps       = 1e-10f;
constexpr float kZFar      = 100.0f;
constexpr float kZRangeInv = 1.0f / (100.0f - 1.0f); // 1/(ZFAR-ZNEAR)
constexpr int   KFRAG      = 8;
constexpr int   TILE       = 256;                    // pixels per workgroup tile
}

// ---------------- CDNA5 Tensor Data Mover staging -------------------------
#if defined(__AMDGCN__) && __has_builtin(__builtin_amdgcn_tensor_load_to_lds) && \
    __has_builtin(__builtin_amdgcn_s_wait_tensorcnt)
#define USE_TDM 1
typedef unsigned int u32x4 __attribute__((ext_vector_type(4)));
typedef int          i32x4 __attribute__((ext_vector_type(4)));
typedef int          i32x8 __attribute__((ext_vector_type(8)));

// Stage `nelem` contiguous f32 from global `gsrc` into LDS byte offset
// `lds_byte` via one TDM descriptor (D#): count=1, type=image(2), data_size=4B,
// single-row 2D tile (tile_dim0 = nelem <= 65535), no pad/iterate/multicast.
// This toolchain exposes the 6-arg builtin form:
//   (u32x4 g0, i32x8 g1, i32x4 g2, i32x4 g3, i32x8 ext, i32 cpol)
// g2/g3/ext are only needed for >2D tensors -> zero-filled here.
__device__ __forceinline__ void tdm_stage(const void* gsrc, unsigned lds_byte,
                                          int nelem) {
  unsigned long long ga = (unsigned long long)(uintptr_t)gsrc;
  u32x4 g0;
  g0[0] = 1u;                                             // count=1 (valid)
  g0[1] = lds_byte;                                       // lds_addr
  g0[2] = (unsigned)(ga & 0xFFFFFFFFu);                   // global_addr[31:0]
  g0[3] = (unsigned)((ga >> 32) & 0x01FFFFFFu) | (2u << 30); // [56:32] | type=2
  i32x8 g1;
  g1[0] = 2 << 16;                        // wg_mask=0, data_size=2 (4 bytes)
  g1[1] = (nelem & 0xFFFF) << 16;         // tensor_dim0[15:0] @ bits 63:48
  g1[2] = (nelem >> 16) | (1 << 16);      // tensor_dim0[31:16], tensor_dim1=1
  g1[3] = (nelem & 0xFFFF) << 16;         // tile_dim0 @ bits 127:112
  g1[4] = 0;                              // tile_dim1=0, tile_dim2=0
  g1[5] = nelem;                          // tensor_dim0_stride[31:0]
  g1[6] = 0;
  g1[7] = 0;
  i32x4 z4 = {0, 0, 0, 0};
  i32x8 z8 = {0, 0, 0, 0, 0, 0, 0, 0};
  __builtin_amdgcn_tensor_load_to_lds(g0, g1, z4, z4, z8, 0);
}
#else
#define USE_TDM 0
#endif

// ---------------- face-normal table precompute ----------------------------
// [F][3][4] f32, 16B-padded rows: 4.8 MB, hot in the 192MB L2 for 8M gathers.
__global__ __launch_bounds__(256)
void fn_precompute_kernel(const float* __restrict__ vn,
                          const int* __restrict__ faces,
                          float* __restrict__ fn, int F) {
  int f = blockIdx.x * blockDim.x + threadIdx.x;
  if (f >= F) return;
  int i0 = faces[3 * f + 0];
  int i1 = faces[3 * f + 1];
  int i2 = faces[3 * f + 2];
  float4 r0 = make_float4(vn[3 * i0 + 0], vn[3 * i0 + 1], vn[3 * i0 + 2], 0.0f);
  float4 r1 = make_float4(vn[3 * i1 + 0], vn[3 * i1 + 1], vn[3 * i1 + 2], 0.0f);
  float4 r2 = make_float4(vn[3 * i2 + 0], vn[3 * i2 + 1], vn[3 * i2 + 2], 0.0f);
  float4* o = reinterpret_cast<float4*>(fn + (size_t)f * 12);
  o[0] = r0; o[1] = r1; o[2] = r2;
}

// ---------------- per-pixel math (shared by all kernels) ------------------
template <bool PRE>
__device__ __forceinline__ float4 blend_math(const float* __restrict__ bw,   // 24
                                             const float* __restrict__ zv,   // 8
                                             const float* __restrict__ dv,   // 8
                                             const int* __restrict__ idx,    // 8
                                             const bool* __restrict__ mask,  // 8
                                             const float* __restrict__ fn,
                                             const float* __restrict__ vn,
                                             const int* __restrict__ faces) {
  float col0[KFRAG], col1[KFRAG], col2[KFRAG];
  float prob[KFRAG], zi[KFRAG];
  float zmax = kEps;
#pragma unroll
  for (int k = 0; k < KFRAG; ++k) {
    float w0 = bw[3 * k + 0], w1 = bw[3 * k + 1], w2 = bw[3 * k + 2];
    float n0, n1, n2;
    if (PRE) {
      const float4* fr = reinterpret_cast<const float4*>(fn + (size_t)idx[k] * 12);
      float4 r0 = fr[0], r1 = fr[1], r2 = fr[2];
      n0 = w0 * r0.x + w1 * r1.x + w2 * r2.x;
      n1 = w0 * r0.y + w1 * r1.y + w2 * r2.y;
      n2 = w0 * r0.z + w1 * r1.z + w2 * r2.z;
    } else {
      int base = idx[k] * 3;
      int i0 = faces[base + 0], i1 = faces[base + 1], i2 = faces[base + 2];
      n0 = w0 * vn[3 * i0 + 0] + w1 * vn[3 * i1 + 0] + w2 * vn[3 * i2 + 0];
      n1 = w0 * vn[3 * i0 + 1] + w1 * vn[3 * i1 + 1] + w2 * vn[3 * i2 + 1];
      n2 = w0 * vn[3 * i0 + 2] + w1 * vn[3 * i1 + 2] + w2 * vn[3 * i2 + 2];
    }
    if (!mask[k]) { n0 = 0.0f; n1 = 0.0f; n2 = 0.0f; }
    col0[k] = (n0 + 1.0f) * 0.5f;   // flip (1,-1,-1), map [-1,1]->[0,1]
    col1[k] = (1.0f - n1) * 0.5f;
    col2[k] = (1.0f - n2) * 0.5f;
    prob[k] = mask[k] ? 1.0f / (1.0f + __expf(dv[k] * kSigmaInv)) : 0.0f;
    zi[k]   = mask[k] ? (kZFar - zv[k]) * kZRangeInv : 0.0f;
    zmax    = fmaxf(zmax, zi[k]);
  }
  float delta = fmaxf(__expf((kEps - zmax) * kGammaInv), kEps);
  float denom = delta;
  float r = 0.0f, g = 0.0f, b = 0.0f, keep = 1.0f;
#pragma unroll
  for (int k = 0; k < KFRAG; ++k) {
    float w = prob[k] * __expf((zi[k] - zmax) * kGammaInv);
    denom += w;
    r += w * col0[k];
    g += w * col1[k];
    b += w * col2[k];
    keep *= (1.0f - prob[k]);
  }
  float inv = 1.0f / denom;
  float bgw = delta * inv;            // BG_COLOR = (1,1,1)
  return make_float4(r * inv + bgw, g * inv + bgw, b * inv + bgw, keep);
}

// ---------------- fallback: fully direct-load kernel ----------------------
template <bool PRE>
__global__ __launch_bounds__(256)
void shade_blend_kernel(const float* __restrict__ vn,
                        const float* __restrict__ bary,
                        const float* __restrict__ zbuf,
                        const float* __restrict__ dists,
                        const int* __restrict__ faces,
                        const int* __restrict__ p2f,
                        const float* __restrict__ fn,
                        float* __restrict__ out, int npix) {
  int pix = blockIdx.x * blockDim.x + threadIdx.x;
  if (pix >= npix) return;
  const int4* p4 = reinterpret_cast<const int4*>(p2f + (size_t)pix * KFRAG);
  int4 pa = p4[0], pb = p4[1];
  int idx[KFRAG] = {pa.x, pa.y, pa.z, pa.w, pb.x, pb.y, pb.z, pb.w};
  bool mask[KFRAG];
#pragma unroll
  for (int k = 0; k < KFRAG; ++k) {
    mask[k] = idx[k] >= 0;
    idx[k]  = mask[k] ? idx[k] : 0;
  }
  if (PRE) {
#pragma unroll
    for (int k = 0; k < KFRAG; ++k)
      __builtin_prefetch(fn + (size_t)idx[k] * 12, 0, 3);
  }
  const float4* b4 = reinterpret_cast<const float4*>(bary + (size_t)pix * (KFRAG * 3));
  float bw[KFRAG * 3];
#pragma unroll
  for (int i = 0; i < 6; ++i) {
    float4 v = b4[i];
    bw[4 * i + 0] = v.x; bw[4 * i + 1] = v.y;
    bw[4 * i + 2] = v.z; bw[4 * i + 3] = v.w;
  }
  const float4* z4 = reinterpret_cast<const float4*>(zbuf + (size_t)pix * KFRAG);
  const float4* d4 = reinterpret_cast<const float4*>(dists + (size_t)pix * KFRAG);
  float4 za = z4[0], zb = z4[1];
  float4 da = d4[0], db = d4[1];
  float zv[KFRAG] = {za.x, za.y, za.z, za.w, zb.x, zb.y, zb.z, zb.w};
  float dv[KFRAG] = {da.x, da.y, da.z, da.w, db.x, db.y, db.z, db.w};
  float4 o = blend_math<PRE>(bw, zv, dv, idx, mask, fn, vn, faces);
  *reinterpret_cast<float4*>(out + (size_t)pix * 4) = o;
}

// ---------------- TDM-staged, double-buffered streaming kernel ------------
// bary (96MB, half the input traffic) is DMA'd tile-by-tile into LDS by the
// Tensor Data Mover while the wave's VMEM path does the latency-critical
// face-normal gathers. p2f/zbuf/dists loads + fn prefetches are issued before
// the s_wait_tensorcnt so they overlap the DMA.
__global__ __launch_bounds__(TILE)
void shade_blend_tdm_kernel(const float* __restrict__ bary,
                            const float* __restrict__ zbuf,
                            const float* __restrict__ dists,
                            const int* __restrict__ p2f,
                            const float* __restrict__ fn,
                            float* __restrict__ out, int ntiles) {
#if USE_TDM
  __shared__ float sb[2][TILE * KFRAG * 3];   // 2 x 24KB double buffer
#endif
  const int tid    = threadIdx.x;
  const int stride = gridDim.x;
  const int t0     = blockIdx.x;
#if USE_TDM
  const bool issuer = (tid >> 5) == 0;        // wave 0 only (TDM ignores EXEC)
  if (issuer && t0 < ntiles)
    tdm_stage(bary + (size_t)t0 * (TILE * KFRAG * 3),
              (unsigned)(uintptr_t)&sb[0][0], TILE * KFRAG * 3);
#endif
  int buf = 0;
  for (int t = t0; t < ntiles; t += stride, buf ^= 1) {
    int pix = t * TILE + tid;

    // ---- direct loads that overlap the TDM wait ----
    const int4* p4 = reinterpret_cast<const int4*>(p2f + (size_t)pix * KFRAG);
    int4 pa = p4[0], pb = p4[1];
    int idx[KFRAG] = {pa.x, pa.y, pa.z, pa.w, pb.x, pb.y, pb.z, pb.w};
    bool mask[KFRAG];
#pragma unroll
    for (int k = 0; k < KFRAG; ++k) {
      mask[k] = idx[k] >= 0;
      idx[k]  = mask[k] ? idx[k] : 0;
    }
#pragma unroll
    for (int k = 0; k < KFRAG; ++k)
      __builtin_prefetch(fn + (size_t)idx[k] * 12, 0, 3);
    const float4* z4 = reinterpret_cast<const float4*>(zbuf + (size_t)pix * KFRAG);
    const float4* d4 = reinterpret_cast<const float4*>(dists + (size_t)pix * KFRAG);
    float4 za = z4[0], zb = z4[1];
    float4 da = d4[0], db = d4[1];
    float zv[KFRAG] = {za.x, za.y, za.z, za.w, zb.x, zb.y, zb.z, zb.w};
    float dv[KFRAG] = {da.x, da.y, da.z, da.w, db.x, db.y, db.z, db.w};

    float bw[KFRAG * 3];
#if USE_TDM
    {
      int tn = t + stride;
      if (issuer) {
        if (tn < ntiles) {
          tdm_stage(bary + (size_t)tn * (TILE * KFRAG * 3),
                    (unsigned)(uintptr_t)&sb[buf ^ 1][0], TILE * KFRAG * 3);
          __builtin_amdgcn_s_wait_tensorcnt(1);   // current tile landed
        } else {
          __builtin_amdgcn_s_wait_tensorcnt(0);
        }
      }
      __syncthreads();
      const float4* lb = reinterpret_cast<const float4*>(&sb[buf][tid * (KFRAG * 3)]);
#pragma unroll
      for (int i = 0; i < 6; ++i) {             // 6x ds_load_b128
        float4 v = lb[i];
        bw[4 * i + 0] = v.x; bw[4 * i + 1] = v.y;
        bw[4 * i + 2] = v.z; bw[4 * i + 3] = v.w;
      }
    }
#else
    {
      const float4* b4 = reinterpret_cast<const float4*>(bary + (size_t)pix * (KFRAG * 3));
#pragma unroll
      for (int i = 0; i < 6; ++i) {
        float4 v = b4[i];
        bw[4 * i + 0] = v.x; bw[4 * i + 1] = v.y;
        bw[4 * i + 2] = v.z; bw[4 * i + 3] = v.w;
      }
    }
#endif
    float4 o = blend_math<true>(bw, zv, dv, idx, mask, fn, nullptr, nullptr);
    *reinterpret_cast<float4*>(out + (size_t)pix * 4) = o;
#if USE_TDM
    __syncthreads();   // all readers done before buf is re-staged next iter
#endif
  }
}

extern "C" void kernel_launch(void* const* d_in, const int* in_sizes, int n_in,
                              void* d_out, int out_size, void* d_ws, size_t ws_size,
                              hipStream_t stream) {
  // setup_inputs() order: verts_normals, bary_coords, zbuf, dists, faces, pix_to_face
  const float* vn    = (const float*)d_in[0];
  const float* bary  = (const float*)d_in[1];
  const float* zbuf  = (const float*)d_in[2];
  const float* dists = (const float*)d_in[3];
  const int*   faces = (const int*)d_in[4];
  const int*   p2f   = (const int*)d_in[5];
  float* out = (float*)d_out;

  int F    = in_sizes[4] / 3;      // faces is [F,3]
  int npix = in_sizes[2] / KFRAG;  // zbuf is [N,H,W,K]

  size_t fn_bytes = (size_t)F * 12 * sizeof(float);
  dim3 blk(256);
  if (ws_size >= fn_bytes) {
    float* fn = (float*)d_ws;
    fn_precompute_kernel<<<dim3((F + 255) / 256), blk, 0, stream>>>(vn, faces, fn, F);
    if (npix % TILE == 0) {
      int ntiles = npix / TILE;
      int grid   = ntiles < 1024 ? ntiles : 1024;   // >=2 tiles/block -> pipelined
      shade_blend_tdm_kernel<<<dim3(grid), dim3(TILE), 0, stream>>>(
          bary, zbuf, dists, p2f, fn, out, ntiles);
    } else {
      shade_blend_kernel<true><<<dim3((npix + 255) / 256), blk, 0, stream>>>(
          vn, bary, zbuf, dists, faces, p2f, fn, out, npix);
    }
  } else {
    shade_blend_kernel<false><<<dim3((npix + 255) / 256), blk, 0, stream>>>(
        vn, bary, zbuf, dists, faces, p2f, nullptr, out, npix);
  }
}